// AttentivePooling_49203145343717
// MI455X (gfx1250) — compile-verified
//
#include <hip/hip_runtime.h>
#include <math.h>

#define B_   2
#define T_   4
#define N2   8192
#define C_   128
#define M_   1024
#define KK   9
#define NH   8
#define DH   64
#define N1   1536      // qkv width = 3*512
#define LDA  132       // padded K for gemm1 (130 -> 132)
#define KT1  33        // 132/4 K-tiles
#define LDQ  1536
#define LDO  512
#define K1   130       // c+2

typedef __attribute__((ext_vector_type(2))) float v2f;
typedef __attribute__((ext_vector_type(8))) float v8f;
typedef __attribute__((ext_vector_type(4))) int   v4i;

#define AS1 __attribute__((address_space(1)))
#define AS3 __attribute__((address_space(3)))

#if defined(__has_builtin)
#if __has_builtin(__builtin_amdgcn_global_load_async_to_lds_b128) && \
    __has_builtin(__builtin_amdgcn_s_wait_asynccnt)
#define USE_ASYNC_LDS 1
#endif
#endif

// ---------------- Kernel 1: farthest point sampling (per batch) ----------------
#define FPS_THREADS 1024
__global__ __launch_bounds__(FPS_THREADS) void fps_kernel(
    const float* __restrict__ pos, int* __restrict__ pivot_idx,
    float* __restrict__ pivot_pos, float* __restrict__ out_tail) {
  __shared__ float dists[N2];
  __shared__ float rval[FPS_THREADS];
  __shared__ int   ridx[FPS_THREADS];
  __shared__ int   s_last;
  const int b = blockIdx.x;
  const int tid = threadIdx.x;
  const float* pb = pos + (size_t)b * N2 * 2;

  for (int i = tid; i < N2; i += FPS_THREADS) dists[i] = __builtin_inff();
  if (tid == 0) { s_last = 0; pivot_idx[b * M_ + 0] = 0; }
  __syncthreads();

  for (int step = 1; step < M_; ++step) {
    const int last = s_last;
    const float px = pb[last * 2 + 0], py = pb[last * 2 + 1];
    float bestv = -__builtin_inff();
    int   besti = 0x7fffffff;
    for (int i = tid; i < N2; i += FPS_THREADS) {
      float dx = pb[i * 2 + 0] - px;
      float dy = pb[i * 2 + 1] - py;
      float d  = dx * dx + dy * dy;
      float dm = fminf(dists[i], d);
      dists[i] = dm;
      if (dm > bestv) { bestv = dm; besti = i; }   // strided ascending -> first max kept
    }
    rval[tid] = bestv; ridx[tid] = besti;
    __syncthreads();
    for (int s = FPS_THREADS / 2; s > 0; s >>= 1) {
      if (tid < s) {
        float v2 = rval[tid + s]; int i2 = ridx[tid + s];
        if (v2 > rval[tid] || (v2 == rval[tid] && i2 < ridx[tid])) {
          rval[tid] = v2; ridx[tid] = i2;
        }
      }
      __syncthreads();
    }
    if (tid == 0) { s_last = ridx[0]; pivot_idx[b * M_ + step] = ridx[0]; }
    __syncthreads();
  }
  for (int i = tid; i < M_; i += FPS_THREADS) {
    int j = pivot_idx[b * M_ + i];
    float xx = pb[j * 2 + 0], yy = pb[j * 2 + 1];
    pivot_pos[(b * M_ + i) * 2 + 0] = xx;
    pivot_pos[(b * M_ + i) * 2 + 1] = yy;
    out_tail[(b * M_ + i) * 2 + 0]  = xx;
    out_tail[(b * M_ + i) * 2 + 1]  = yy;
  }
}

// ---------------- Kernel 2: top-9 nearest neighbors per pivot ----------------
#define KNN_THREADS 256
__global__ __launch_bounds__(KNN_THREADS) void knn_kernel(
    const float* __restrict__ pos, const float* __restrict__ pivot_pos,
    int* __restrict__ nbr) {
  __shared__ float sd[KNN_THREADS * KK];
  __shared__ int   si[KNN_THREADS * KK];
  const int bm = blockIdx.x;
  const int b  = bm / M_;
  const int tid = threadIdx.x;
  const float* pb = pos + (size_t)b * N2 * 2;
  const float px = pivot_pos[bm * 2 + 0], py = pivot_pos[bm * 2 + 1];
  const float pp = px * px + py * py;

  float bd[KK]; int bi[KK];
  for (int s = 0; s < KK; ++s) { bd[s] = __builtin_inff(); bi[s] = 0x7fffffff; }
  for (int i = tid; i < N2; i += KNN_THREADS) {
    float qx = pb[i * 2 + 0], qy = pb[i * 2 + 1];
    float qq = qx * qx + qy * qy;
    float dt = px * qx + py * qy;
    float d2 = (pp + qq) - 2.0f * dt;   // matches reference formula order
    if (d2 < bd[KK - 1] || (d2 == bd[KK - 1] && i < bi[KK - 1])) {
      int s = KK - 1;
      while (s > 0 && (d2 < bd[s - 1] || (d2 == bd[s - 1] && i < bi[s - 1]))) {
        bd[s] = bd[s - 1]; bi[s] = bi[s - 1]; --s;
      }
      bd[s] = d2; bi[s] = i;
    }
  }
  for (int s = 0; s < KK; ++s) { sd[tid * KK + s] = bd[s]; si[tid * KK + s] = bi[s]; }
  __syncthreads();
  for (int stride = KNN_THREADS / 2; stride > 0; stride >>= 1) {
    if (tid < stride) {
      float* da = &sd[tid * KK];            int* ia = &si[tid * KK];
      float* db = &sd[(tid + stride) * KK]; int* ib = &si[(tid + stride) * KK];
      float md[KK]; int mi[KK];
      int a = 0, c = 0;
      for (int s = 0; s < KK; ++s) {        // a+c == s <= 8, no OOB
        bool takeA = (da[a] < db[c]) || (da[a] == db[c] && ia[a] <= ib[c]);
        if (takeA) { md[s] = da[a]; mi[s] = ia[a]; ++a; }
        else       { md[s] = db[c]; mi[s] = ib[c]; ++c; }
      }
      for (int s = 0; s < KK; ++s) { da[s] = md[s]; ia[s] = mi[s]; }
    }
    __syncthreads();
  }
  if (tid < KK) nbr[bm * KK + tid] = si[tid];
}

// ---------------- Kernel 3: gather + qkv GEMM + RoPE attention + out GEMM ----------------
__global__ __launch_bounds__(256) void attn_kernel(
    const float* __restrict__ x, const float* __restrict__ pos,
    const float* __restrict__ Wqkv, const float* __restrict__ Wout,
    const float* __restrict__ bout, const float* __restrict__ pivot_pos,
    const int* __restrict__ nbr, float* __restrict__ out) {
  extern __shared__ float smem[];
  float* sA    = smem;                  // 16 x 132 padded xin
  float* sQ    = sA + 16 * LDA;         // 16 x 1536  q|k|v
  float* sO    = sQ + 16 * LDQ;         // 16 x 512   attention output
  float* sRelX = sO + 16 * LDO;         // 16
  float* sRelY = sRelX + 16;            // 16
  int*   sNbr  = (int*)(sRelY + 16);    // 16

  const int g    = blockIdx.x;          // ((b*T + t)*M + m)
  const int m_   = g % M_;
  const int bt   = g / M_;
  const int t_   = bt % T_;
  const int b    = bt / T_;
  const int bm   = b * M_ + m_;
  const int tid  = threadIdx.x;
  const int wave = tid >> 5;            // == head index
  const int lane = tid & 31;

  // prefetch GEMM2 weights into near caches; they are consumed only after
  // GEMM1 + attention, so this hides the L2->WGP latency behind real work.
  __builtin_prefetch(&Wout[(size_t)tid * 256], 0, 0);

  // stage neighbor indices + relative positions
  if (tid < KK) {
    int j = nbr[bm * KK + tid];
    sNbr[tid] = j;
    float pvx = pivot_pos[bm * 2 + 0], pvy = pivot_pos[bm * 2 + 1];
    sRelX[tid] = pos[((size_t)b * N2 + j) * 2 + 0] - pvx;
    sRelY[tid] = pos[((size_t)b * N2 + j) * 2 + 1] - pvy;
  }
  __syncthreads();

  // stage A = xin (rows 0..8 real, rest zero; cols 0..127 = x gather, 128/129 = rel)
#ifdef USE_ASYNC_LDS
  // gathered feature rows move global->LDS on the async path (no VGPR round-trip)
  for (int idx = tid; idx < KK * 32; idx += 256) {
    int r = idx >> 5, seg = idx & 31;
    const float* gsrc = x + (((size_t)(b * T_ + t_)) * N2 + sNbr[r]) * C_ + seg * 4;
    float* ldst = sA + r * LDA + seg * 4;
    __builtin_amdgcn_global_load_async_to_lds_b128(
        (AS1 v4i*)(uintptr_t)(const void*)gsrc,
        (AS3 v4i*)(uintptr_t)(void*)ldst, 0, 0);
  }
  for (int idx = tid; idx < 16 * LDA; idx += 256) {   // rel cols + zero padding
    int r = idx / LDA, cc = idx % LDA;
    if (r < KK && cc < C_) continue;                  // covered by async copies
    float v = 0.0f;
    if (r < KK) {
      if (cc == C_)          v = sRelX[r];
      else if (cc == C_ + 1) v = sRelY[r];
    }
    sA[idx] = v;
  }
  __builtin_amdgcn_s_wait_asynccnt(0);
  __syncthreads();
#else
  for (int idx = tid; idx < 16 * LDA; idx += 256) {
    int r = idx / LDA, cc = idx % LDA;
    float v = 0.0f;
    if (r < KK) {
      if (cc < C_) {
        int j = sNbr[r];
        v = x[(((size_t)(b * T_ + t_)) * N2 + j) * C_ + cc];
      } else if (cc == C_)     v = sRelX[r];
      else if (cc == C_ + 1)   v = sRelY[r];
    }
    sA[idx] = v;
  }
  __syncthreads();
#endif

  // ---- GEMM1: qkv = xin @ Wqkv via V_WMMA_F32_16X16X4_F32 ----
  const int arow  = lane & 15;                 // M row (both lane halves)
  const int akoff = (lane < 16) ? 0 : 2;       // K sub-offset per lane half
  float a0[KT1], a1[KT1];
#pragma unroll
  for (int kt = 0; kt < KT1; ++kt) {
    a0[kt] = sA[arow * LDA + kt * 4 + akoff + 0];
    a1[kt] = sA[arow * LDA + kt * 4 + akoff + 1];
  }
  const int h  = wave;
  const int bn = lane & 15;                    // N column within tile
#pragma unroll 1
  for (int nt = 0; nt < 12; ++nt) {            // 4 tiles each of q,k,v for this head
    int n0 = (nt >> 2) * 512 + h * DH + (nt & 3) * 16;
    v8f acc = {0.f, 0.f, 0.f, 0.f, 0.f, 0.f, 0.f, 0.f};
#pragma unroll
    for (int kt = 0; kt < KT1; ++kt) {
      int k0 = kt * 4 + akoff;
      int ka = (k0     < K1) ? k0     : (K1 - 1);  // clamp; padded A cols are 0
      int kb = (k0 + 1 < K1) ? k0 + 1 : (K1 - 1);
      v2f av; av.x = a0[kt]; av.y = a1[kt];
      v2f bv; bv.x = Wqkv[(size_t)ka * N1 + n0 + bn];
              bv.y = Wqkv[(size_t)kb * N1 + n0 + bn];
      acc = __builtin_amdgcn_wmma_f32_16x16x4_f32(false, av, false, bv,
                                                  (short)0, acc, false, false);
    }
    int rbase = (lane < 16) ? 0 : 8;             // D tile: rows 0-7 / 8-15
#pragma unroll
    for (int rr = 0; rr < 8; ++rr)
      sQ[(rbase + rr) * LDQ + n0 + bn] = acc[rr];
  }
  __syncthreads();

  // ---- 2D RoPE on q and k (wave h handles head h, rows 0..8) ----
  {
    const int p = lane & 15;
    const float invf = powf(10000.0f, -((float)p) / 16.0f);
    const int qc = h * DH;
    const int kc = 512 + h * DH;
    const int d0 = (lane < 16) ? p : (32 + p);   // x-half / y-half
    const int d1 = d0 + 16;
#pragma unroll 1
    for (int r = 0; r < KK; ++r) {
      float coord = (lane < 16) ? sRelX[r] : sRelY[r];
      float f  = coord * 2048.0f * invf;
      float cf = cosf(f), sf = sinf(f);
      float q0 = sQ[r * LDQ + qc + d0], q1 = sQ[r * LDQ + qc + d1];
      sQ[r * LDQ + qc + d0] = q0 * cf - q1 * sf;
      sQ[r * LDQ + qc + d1] = q1 * cf + q0 * sf;
      float k0 = sQ[r * LDQ + kc + d0], k1 = sQ[r * LDQ + kc + d1];
      sQ[r * LDQ + kc + d0] = k0 * cf - k1 * sf;
      sQ[r * LDQ + kc + d1] = k1 * cf + k0 * sf;
    }
  }
  __syncthreads();

  // ---- tiny attention per head (9x9 softmax), write sO rows 0..8, zero 9..15 ----
  {
    const int qc = h * DH, kc = 512 + h * DH, vc = 1024 + h * DH, oc = h * DH;
    for (int r = KK; r < 16; ++r) {
      sO[r * LDO + oc + lane]      = 0.0f;
      sO[r * LDO + oc + lane + 32] = 0.0f;
    }
#pragma unroll 1
    for (int i = 0; i < KK; ++i) {
      int j = (lane < KK) ? lane : 0;
      float dot = 0.0f;
#pragma unroll 16
      for (int d = 0; d < DH; ++d)
        dot += sQ[i * LDQ + qc + d] * sQ[j * LDQ + kc + d];
      dot *= 0.125f;                              // dh^-0.5
      float dj[KK];
#pragma unroll
      for (int j2 = 0; j2 < KK; ++j2) dj[j2] = __shfl(dot, j2, 32);
      float mx = dj[0];
#pragma unroll
      for (int j2 = 1; j2 < KK; ++j2) mx = fmaxf(mx, dj[j2]);
      float ssum = 0.0f;
#pragma unroll
      for (int j2 = 0; j2 < KK; ++j2) { dj[j2] = expf(dj[j2] - mx); ssum += dj[j2]; }
      float inv = 1.0f / ssum;
      float o0 = 0.0f, o1 = 0.0f;
#pragma unroll
      for (int j2 = 0; j2 < KK; ++j2) {
        float w = dj[j2] * inv;
        o0 += w * sQ[j2 * LDQ + vc + lane];
        o1 += w * sQ[j2 * LDQ + vc + lane + 32];
      }
      sO[i * LDO + oc + lane]      = o0;
      sO[i * LDO + oc + lane + 32] = o1;
    }
  }
  __syncthreads();

  // ---- GEMM2: attn_out(16x512) @ Wout(512x128); wave owns one 16-col N-tile ----
  {
    const int n0 = wave * 16;
    v8f acc = {0.f, 0.f, 0.f, 0.f, 0.f, 0.f, 0.f, 0.f};
#pragma unroll 4
    for (int kt = 0; kt < 128; ++kt) {
      int k0 = kt * 4 + akoff;
      v2f av; av.x = sO[arow * LDO + k0];
              av.y = sO[arow * LDO + k0 + 1];
      v2f bv; bv.x = Wout[(size_t)k0 * C_ + n0 + bn];
              bv.y = Wout[(size_t)(k0 + 1) * C_ + n0 + bn];
      acc = __builtin_amdgcn_wmma_f32_16x16x4_f32(false, av, false, bv,
                                                  (short)0, acc, false, false);
    }
    // mean over rows 0..8: lanes<16 hold rows0-7, partner lane+16 holds row8 (rows 9-15 = 0)
    float s = acc[0] + acc[1] + acc[2] + acc[3] + acc[4] + acc[5] + acc[6] + acc[7];
    float other = __shfl_xor(s, 16, 32);
    if (lane < 16) {
      float val = (s + other) * (1.0f / 9.0f) + bout[n0 + bn];
      out[(size_t)g * C_ + n0 + bn] = val;
    }
  }
}

extern "C" void kernel_launch(void* const* d_in, const int* in_sizes, int n_in,
                              void* d_out, int out_size, void* d_ws, size_t ws_size,
                              hipStream_t stream) {
  (void)in_sizes; (void)n_in; (void)out_size; (void)ws_size;
  const float* x    = (const float*)d_in[0];
  const float* pos  = (const float*)d_in[1];
  const float* Wqkv = (const float*)d_in[2];
  const float* Wout = (const float*)d_in[3];
  const float* bout = (const float*)d_in[4];
  float* out = (float*)d_out;

  char* ws = (char*)d_ws;
  int*   pivot_idx = (int*)ws;                                   // B*M ints
  float* pivot_pos = (float*)(ws + (size_t)B_ * M_ * 4);         // B*M*2 floats
  int*   nbr       = (int*)(ws + (size_t)B_ * M_ * 4 + (size_t)B_ * M_ * 2 * 4); // B*M*KK ints

  float* out_tail = out + (size_t)B_ * T_ * M_ * C_;             // pivot_pos output

  fps_kernel<<<B_, FPS_THREADS, 0, stream>>>(pos, pivot_idx, pivot_pos, out_tail);
  knn_kernel<<<B_ * M_, KNN_THREADS, 0, stream>>>(pos, pivot_pos, nbr);

  size_t smem = (size_t)(16 * LDA + 16 * LDQ + 16 * LDO + 16 + 16 + 16) * 4;
  attn_kernel<<<B_ * T_ * M_, 256, smem, stream>>>(x, pos, Wqkv, Wout, bout,
                                                   pivot_pos, nbr, out);
}